// GraphConvolutionLayer_2396591751760
// MI455X (gfx1250) — compile-verified
//
#include <hip/hip_runtime.h>

#define N_NODES 10000
#define N_EDGES 640000
#define IN_F    128
#define OUT_F   256
#define M_TILES (N_NODES / 16)   // 625
#define N_TILES (OUT_F / 16)     // 16
#define EDGES_PER_BLOCK 512
#define WAVES_PER_BLOCK 8

static_assert(N_EDGES % EDGES_PER_BLOCK == 0, "full chunks only");
static_assert(N_NODES % 16 == 0, "exact M tiling");

typedef float v2f __attribute__((ext_vector_type(2)));
typedef float v8f __attribute__((ext_vector_type(8)));

// ---------------------------------------------------------------------------
// Stage 0: zero the h accumulator in workspace (d_ws is poisoned once and we
// accumulate with atomics, so every launch must re-zero).
// ---------------------------------------------------------------------------
__global__ void zero_h_kernel(float* __restrict__ h) {
    int i = blockIdx.x * blockDim.x + threadIdx.x;
    if (i < N_NODES * IN_F) h[i] = 0.0f;
}

// ---------------------------------------------------------------------------
// Stage 1: gather + segment-sum exploiting sorted dst.
//  * 128 threads = one per feature column -> each edge is one coalesced 512B
//    row load of x.
//  * Chunk indices staged once into LDS (coalesced), so the serial edge loop
//    reads {src,dst} with a single broadcast ds_load_b64 instead of two
//    redundant block-uniform global loads per edge.
//  * Runs of equal dst accumulate in registers; global_atomic_add_f32 only at
//    segment boundaries (avg run length = 64 edges) and at chunk edges.
//  * global_prefetch of the row 16 edges ahead, once per 4-edge group.
// ---------------------------------------------------------------------------
__global__ __launch_bounds__(IN_F)
void gather_segsum_kernel(const float* __restrict__ x,
                          const int*   __restrict__ src,
                          const int*   __restrict__ dst,
                          float*       __restrict__ h) {
    __shared__ int2 idx[EDGES_PER_BLOCK];             // 4 KB: {src, dst} per edge

    const int f  = threadIdx.x;                       // feature 0..127
    const int e0 = blockIdx.x * EDGES_PER_BLOCK;

    // Stage indices (coalesced; 4 edges per thread).
    for (int i = f; i < EDGES_PER_BLOCK; i += IN_F) {
        int2 p;
        p.x = src[e0 + i];
        p.y = dst[e0 + i];
        idx[i] = p;
    }
    __syncthreads();

    int   prev = idx[0].y;                            // block-uniform
    float acc  = 0.0f;

#pragma unroll 4
    for (int i = 0; i < EDGES_PER_BLOCK; ++i) {
        const int2 p = idx[i];                        // broadcast ds_load_b64
        if (p.y != prev) {                            // uniform branch
            atomicAdd(&h[prev * IN_F + f], acc);      // flush finished segment
            acc  = 0.0f;
            prev = p.y;
        }
        acc += x[p.x * IN_F + f];                     // coalesced 512B row
        if ((i & 3) == 0 && i + 16 < EDGES_PER_BLOCK) {
            __builtin_prefetch(&x[idx[i + 16].x * IN_F + f], 0, 0);
        }
    }
    atomicAdd(&h[prev * IN_F + f], acc);              // chunk-boundary flush
}

// ---------------------------------------------------------------------------
// Stage 2: out = h @ W + b using V_WMMA_F32_16X16X4_F32 (exact f32 path).
// Block = 8 waves; each wave owns one 16x16 output tile (8 M-tiles x 1 N-tile
// per block). The 128x16 W panel for this N-tile is staged once in LDS.
//
// Per ISA VGPR layouts (wave32):
//   A 16x4 f32 : lane half = K pair selector; A.x = K=2*half, A.y = K=2*half+1,
//                M = lane & 15.
//   B 4x16 f32 : mirrored; N = lane & 15.
//   C/D 16x16  : VGPR r -> M = r + 8*half, N = lane & 15.
// ---------------------------------------------------------------------------
__global__ __launch_bounds__(32 * WAVES_PER_BLOCK)
void wmma_gemm_bias_kernel(const float* __restrict__ h,
                           const float* __restrict__ W,
                           const float* __restrict__ b,
                           float*       __restrict__ out) {
    __shared__ float Wlds[IN_F][16];                  // 8 KB panel W[:, n0:n0+16]

    const int tid   = threadIdx.x;
    const int wave  = tid >> 5;
    const int lane  = tid & 31;
    const int laneN = lane & 15;
    const int half  = lane >> 4;

    const int n0 = blockIdx.y * 16;

    // Stage W panel: 2048 floats, 256 threads -> 8 each, coalesced.
    for (int i = tid; i < IN_F * 16; i += 32 * WAVES_PER_BLOCK) {
        const int k = i >> 4;
        const int n = i & 15;
        Wlds[k][n] = W[k * OUT_F + n0 + n];
    }
    __syncthreads();

    const int mtile = blockIdx.x * WAVES_PER_BLOCK + wave;
    if (mtile >= M_TILES) return;                     // wave-uniform; EXEC stays full
    const int m0 = mtile * 16;

    // Bias depends only on N -> identical across all 8 C VGPRs.
    const float bias = b[n0 + laneN];
    v8f c;
#pragma unroll
    for (int r = 0; r < 8; ++r) c[r] = bias;

    const float* arow = h + (m0 + laneN) * IN_F;      // A row for this lane (M = lane&15)

#pragma unroll 8
    for (int k = 0; k < IN_F; k += 4) {
        const v2f a = *(const v2f*)(arow + k + 2 * half);   // 8B aligned contiguous pair
        v2f bb;
        bb[0] = Wlds[k + 2 * half][laneN];            // lane halves hit disjoint banks
        bb[1] = Wlds[k + 2 * half + 1][laneN];
        // 8 args: (neg_a, A, neg_b, B, c_mod, C, reuse_a, reuse_b)
        c = __builtin_amdgcn_wmma_f32_16x16x4_f32(
                false, a, false, bb, (short)0, c, false, false);
    }

    // D layout: VGPR r -> row m0 + r + 8*half, col n0 + laneN.
#pragma unroll
    for (int r = 0; r < 8; ++r) {
        out[(m0 + r + 8 * half) * OUT_F + n0 + laneN] = c[r];
    }
}

// ---------------------------------------------------------------------------
extern "C" void kernel_launch(void* const* d_in, const int* in_sizes, int n_in,
                              void* d_out, int out_size, void* d_ws, size_t ws_size,
                              hipStream_t stream) {
    const float* x   = (const float*)d_in[0];   // [N_NODES, IN_F]
    const int*   src = (const int*)  d_in[1];   // [N_EDGES]
    const int*   dst = (const int*)  d_in[2];   // [N_EDGES] (sorted)
    const float* W   = (const float*)d_in[3];   // [IN_F, OUT_F]
    const float* b   = (const float*)d_in[4];   // [OUT_F]
    float*       out = (float*)d_out;           // [N_NODES, OUT_F]
    float*       h   = (float*)d_ws;            // [N_NODES, IN_F] scratch (5.12 MB)

    // Stage 0: zero h
    {
        const int n = N_NODES * IN_F;
        zero_h_kernel<<<(n + 255) / 256, 256, 0, stream>>>(h);
    }
    // Stage 1: gather + segment-sum (dst sorted -> register runs + boundary atomics)
    {
        const int blocks = N_EDGES / EDGES_PER_BLOCK; // 1250
        gather_segsum_kernel<<<blocks, IN_F, 0, stream>>>(x, src, dst, h);
    }
    // Stage 2: WMMA GEMM + bias
    {
        dim3 grid((M_TILES + WAVES_PER_BLOCK - 1) / WAVES_PER_BLOCK, N_TILES); // (79, 16)
        wmma_gemm_bias_kernel<<<grid, 32 * WAVES_PER_BLOCK, 0, stream>>>(h, W, b, out);
    }
}